// LightGCN_2_63144609186443
// MI455X (gfx1250) — compile-verified
//
#include <hip/hip_runtime.h>
#include <hip/hip_bf16.h>
#include <stdint.h>

// LightGCN propagation on MI455X (gfx1250).
// Bandwidth/atomic-bound SpMM; working set (~90MB) fits in 192MB L2.
// CDNA5 paths: double-buffered global_load_async_to_lds_b128 gather pipeline
// (ASYNCcnt in-order completion lets `s_wait_asynccnt <= WPT` retire the old
// buffer while the new one is in flight), s_wait_asynccnt, global_prefetch_b8,
// no-return global_atomic_add_f32 scatters.

#define DIM        64
#define CHUNKS     16          // DIM/4 float4 chunks per embedding row
#define TILE_E     128         // edges staged per buffer
#define THREADS    256
#define WPT        8           // (TILE_E*CHUNKS)/THREADS async gathers per thread

__global__ void zero_kernel(float* __restrict__ p, int n4) {
    int i = blockIdx.x * blockDim.x + threadIdx.x;
    float4 z = make_float4(0.f, 0.f, 0.f, 0.f);
    for (; i < n4; i += gridDim.x * blockDim.x) ((float4*)p)[i] = z;
}

__global__ void copy_kernel(float* __restrict__ dst, const float* __restrict__ src, int n4) {
    int i = blockIdx.x * blockDim.x + threadIdx.x;
    if (i < n4) ((float4*)dst)[i] = ((const float4*)src)[i];
}

__global__ void acc_kernel(float* __restrict__ dst, const float* __restrict__ src, int n4) {
    int i = blockIdx.x * blockDim.x + threadIdx.x;
    if (i < n4) {
        float4 a = ((float4*)dst)[i];
        float4 b = ((const float4*)src)[i];
        a.x += b.x; a.y += b.y; a.z += b.z; a.w += b.w;
        ((float4*)dst)[i] = a;
    }
}

__global__ void scale_kernel(float* __restrict__ p, float s, int n4) {
    int i = blockIdx.x * blockDim.x + threadIdx.x;
    if (i < n4) {
        float4 a = ((float4*)p)[i];
        a.x *= s; a.y *= s; a.z *= s; a.w *= s;
        ((float4*)p)[i] = a;
    }
}

__global__ void deg_kernel(const int* __restrict__ row, const float* __restrict__ ew,
                           float* __restrict__ deg, int E) {
    int e = blockIdx.x * blockDim.x + threadIdx.x;
    if (e < E) atomicAdd(&deg[row[e]], ew[e]);
}

__global__ void dinv_kernel(float* __restrict__ deg, int n) {
    int i = blockIdx.x * blockDim.x + threadIdx.x;
    if (i < n) {
        float d = deg[i];
        deg[i] = (d > 0.f) ? rsqrtf(d) : 0.f;
    }
}

__global__ void edgew_kernel(const int* __restrict__ row, const int* __restrict__ col,
                             const float* __restrict__ ew, const float* __restrict__ dinv,
                             float* __restrict__ w, int E) {
    int e = blockIdx.x * blockDim.x + threadIdx.x;
    if (e < E) w[e] = dinv[row[e]] * ew[e] * dinv[col[e]];
}

// Issue one tile's worth of lane-addressed async gathers: 16 lanes cover one
// 256B embedding row; each thread issues WPT b128 gathers (ASYNCcnt-tracked,
// zero VGPR cost while in flight). Tail edges are clamped so EXEC stays full.
__device__ __forceinline__ void issue_tile(const int* __restrict__ col,
                                           const float* __restrict__ x,
                                           int e0, int E, unsigned ldsBase, int t) {
#pragma unroll
    for (int i = 0; i < WPT; ++i) {
        const int chunk = t + i * THREADS;            // 0..TILE_E*CHUNKS-1
        const int el = chunk >> 4;                    // edge within tile
        const int c  = chunk & (CHUNKS - 1);          // float4 chunk within row
        const int e  = e0 + el;
        const int cc = col[min(e, E - 1)];
        const uint64_t g = (uint64_t)(uintptr_t)(x + (size_t)cc * DIM + (size_t)c * 4);
        const unsigned l = ldsBase + (unsigned)chunk * 16u;
        asm volatile("global_load_async_to_lds_b128 %0, %1, off"
                     :: "v"(l), "v"(g) : "memory");
    }
}

// y[row[e], :] += w[e] * x[col[e], :]  — software-pipelined:
//   prologue: gather tile0 -> A
//   loop:     gather tile(k+1) -> other buffer, s_wait_asynccnt <= WPT
//             (in-order completion => current buffer fully landed),
//             barrier, scale + atomic-scatter current buffer, barrier.
__global__ void spmm_async_kernel(const int* __restrict__ row, const int* __restrict__ col,
                                  const float* __restrict__ w, const float* __restrict__ x,
                                  float* __restrict__ y, int E) {
    __shared__ float4 tileA[TILE_E * CHUNKS];   // 32 KB
    __shared__ float4 tileB[TILE_E * CHUNKS];   // 32 KB
    const int t = threadIdx.x;
    const int nTiles = (E + TILE_E - 1) / TILE_E;
    // Low 32 bits of a flat pointer into LDS == byte offset in LDS space.
    const unsigned baseA = (unsigned)(uintptr_t)(void*)&tileA[0];
    const unsigned baseB = (unsigned)(uintptr_t)(void*)&tileB[0];

    int tb = blockIdx.x;
    if (tb >= nTiles) return;

    issue_tile(col, x, tb * TILE_E, E, baseA, t);

    int parity = 0;
    for (; tb < nTiles; tb += gridDim.x) {
        const int nxt = tb + gridDim.x;
        if (nxt < nTiles) {
            // Speculative prefetch of the next tile's edge metadata.
            __builtin_prefetch(col + nxt * TILE_E, 0, 0);
            __builtin_prefetch(w   + nxt * TILE_E, 0, 0);
            issue_tile(col, x, nxt * TILE_E, E, parity ? baseA : baseB, t);
            // WPT newest (next buffer) may remain in flight; the WPT oldest
            // (current buffer) are guaranteed retired at <= WPT.
            asm volatile("s_wait_asynccnt %0" :: "i"(WPT) : "memory");
        } else {
            asm volatile("s_wait_asynccnt 0" ::: "memory");
        }
        __syncthreads();   // all waves' gathers into current buffer visible

        const float4* __restrict__ cur = parity ? tileB : tileA;
        const int e0 = tb * TILE_E;
#pragma unroll
        for (int i = 0; i < WPT; ++i) {
            const int chunk = t + i * THREADS;
            const int el = chunk >> 4;
            const int c  = chunk & (CHUNKS - 1);
            const int e  = e0 + el;
            const bool valid = (e < E);
            const float wv = valid ? w[e] : 0.0f;     // tail contributes 0
            const int  r   = valid ? row[e] : 0;
            const float4 v = cur[chunk];
            float* dst = y + (size_t)r * DIM + (size_t)c * 4;
            atomicAdd(dst + 0, v.x * wv);
            atomicAdd(dst + 1, v.y * wv);
            atomicAdd(dst + 2, v.z * wv);
            atomicAdd(dst + 3, v.w * wv);
        }
        __syncthreads();   // current buffer free for reuse by next issue
        parity ^= 1;
    }
}

extern "C" void kernel_launch(void* const* d_in, const int* in_sizes, int n_in,
                              void* d_out, int out_size, void* d_ws, size_t ws_size,
                              hipStream_t stream) {
    const int* edge_index = (const int*)d_in[0];     // [2, E] flat: row then col
    const float* ew       = (const float*)d_in[1];   // [E]
    const float* embs     = (const float*)d_in[2];   // [N, D]
    const int E = in_sizes[1];
    const int N = in_sizes[2] / DIM;

    const int* row = edge_index;
    const int* col = edge_index + E;

    // Workspace layout (floats): deg[N] | w[E] | xA[N*D] | xB[N*D]
    float* deg = (float*)d_ws;
    float* w   = deg + N;
    float* xA  = w + E;
    float* xB  = xA + (size_t)N * DIM;
    float* out = (float*)d_out;

    const int nd4  = (N * DIM) / 4;
    const int eb   = (E + THREADS - 1) / THREADS;
    const int nb   = (N + THREADS - 1) / THREADS;
    const int nd4b = (nd4 + THREADS - 1) / THREADS;
    const int n4deg = (N + 3) / 4;
    const int tiles = (E + TILE_E - 1) / TILE_E;
    const int spmmBlocks = min(tiles, 4096);   // several tiles/block -> pipeline pays

    // 1) normalization weights: w = dinv[row] * ew * dinv[col]
    zero_kernel<<<(n4deg + THREADS - 1) / THREADS, THREADS, 0, stream>>>(deg, n4deg);
    deg_kernel<<<eb, THREADS, 0, stream>>>(row, ew, deg, E);
    dinv_kernel<<<nb, THREADS, 0, stream>>>(deg, N);
    edgew_kernel<<<eb, THREADS, 0, stream>>>(row, col, ew, deg, w, E);

    // 2) acc = embs
    copy_kernel<<<nd4b, THREADS, 0, stream>>>(out, embs, nd4);

    // 3) three propagation hops, acc += x each time
    zero_kernel<<<nd4b, THREADS, 0, stream>>>(xA, nd4);
    spmm_async_kernel<<<spmmBlocks, THREADS, 0, stream>>>(row, col, w, embs, xA, E);
    acc_kernel<<<nd4b, THREADS, 0, stream>>>(out, xA, nd4);

    zero_kernel<<<nd4b, THREADS, 0, stream>>>(xB, nd4);
    spmm_async_kernel<<<spmmBlocks, THREADS, 0, stream>>>(row, col, w, xA, xB, E);
    acc_kernel<<<nd4b, THREADS, 0, stream>>>(out, xB, nd4);

    zero_kernel<<<nd4b, THREADS, 0, stream>>>(xA, nd4);
    spmm_async_kernel<<<spmmBlocks, THREADS, 0, stream>>>(row, col, w, xB, xA, E);
    acc_kernel<<<nd4b, THREADS, 0, stream>>>(out, xA, nd4);

    // 4) mean over {x0..x3}
    scale_kernel<<<nd4b, THREADS, 0, stream>>>(out, 0.25f, nd4);
}